// GCNBlock_17540646437112
// MI455X (gfx1250) — compile-verified
//
#include <hip/hip_runtime.h>

// GCN block: h = x@W (fp32 WMMA), scatter-aggregate over pos edges + self loops,
// BatchNorm(training, biased var) + ReLU. Neg edges have weight 0 -> fully dead.
//
// Sizes fixed by the reference.
#define N_NODES 50000
#define D 128          // D_IN == D_OUT == 128
#define E_POS 500000
#define BN_EPS 1e-5f

typedef __attribute__((ext_vector_type(2))) float v2f;
typedef __attribute__((ext_vector_type(4))) float v4f;
typedef __attribute__((ext_vector_type(8))) float v8f;

// ---------------------------------------------------------------------------
// 1) Per-call init: out = b (bias folded in; aggregation atomically adds on
//    top), deg = 1.0 (self-loop weight), stats = 0.
// ---------------------------------------------------------------------------
__global__ void init_kernel(float* __restrict__ out, const float* __restrict__ b,
                            float* __restrict__ deg, float* __restrict__ stats) {
  int tid = blockIdx.x * blockDim.x + threadIdx.x;
  if (tid < N_NODES * D) out[tid] = b[tid & (D - 1)];
  if (tid < N_NODES)     deg[tid] = 1.0f;
  if (tid < 2 * D)       stats[tid] = 0.0f;
}

// ---------------------------------------------------------------------------
// 2) deg[col] += 1 for each positive edge (neg edges carry weight 0 -> skip).
//    pos_ei layout: row = pos_ei[e], col = pos_ei[E_POS + e].
// ---------------------------------------------------------------------------
__global__ void deg_kernel(const int* __restrict__ pos_ei, float* __restrict__ deg) {
  int e = blockIdx.x * blockDim.x + threadIdx.x;
  if (e < E_POS) atomicAdd(&deg[pos_ei[E_POS + e]], 1.0f);
}

// ---------------------------------------------------------------------------
// 3) dis = rsqrt(deg)  (deg >= 1 always, so the deg>0 guard is vacuous)
// ---------------------------------------------------------------------------
__global__ void rsqrt_kernel(float* __restrict__ deg) {
  int n = blockIdx.x * blockDim.x + threadIdx.x;
  if (n < N_NODES) deg[n] = rsqrtf(deg[n]);
}

// ---------------------------------------------------------------------------
// 4) h = x @ W  with V_WMMA_F32_16X16X4_F32, one wave32 per 16x16 tile.
//    ISA layouts (32-bit):
//      A 16x4 : lane L -> M=L%16 ; VGPR{0,1} -> K=(L/16)*2+{0,1}
//      B 4x16 : lane L -> N=L%16 ; VGPR{0,1} -> K=(L/16)*2+{0,1}
//      C 16x16: VGPR r -> M=r+(L/16)*8 ; N=L%16
//    25000 tiles exactly; no divergence -> EXEC all 1s as WMMA requires.
// ---------------------------------------------------------------------------
__global__ void gemm_wmma_kernel(const float* __restrict__ x,
                                 const float* __restrict__ W,
                                 float* __restrict__ h) {
  const int lane = threadIdx.x & 31;
  const int wave = threadIdx.x >> 5;
  const int tile = blockIdx.x * 4 + wave;   // 4 waves / 128-thread block
  const int mt   = tile >> 3;               // 8 N-tiles per M-tile (D/16 == 8)
  const int nt   = tile & 7;
  const int m0   = mt * 16;
  const int n0   = nt * 16;
  const int half = lane >> 4;               // 0: K lo pair, 1: K hi pair
  const int l16  = lane & 15;

  const float* arow = x + (size_t)(m0 + l16) * D;  // A: row m, contiguous K
  const float* bcol = W + (n0 + l16);              // B: column n, stride-D K

  v8f c = {0.f, 0.f, 0.f, 0.f, 0.f, 0.f, 0.f, 0.f};
#pragma unroll
  for (int k0 = 0; k0 < D; k0 += 4) {
    const int k = k0 + half * 2;
    v2f a = *(const v2f*)(arow + k);   // 8B-aligned (k even)
    v2f b;
    b.x = bcol[(size_t)(k    ) * D];
    b.y = bcol[(size_t)(k + 1) * D];
    // 8 args: (neg_a, A, neg_b, B, c_mod, C, reuse_a, reuse_b)
    c = __builtin_amdgcn_wmma_f32_16x16x4_f32(false, a, false, b,
                                              (short)0, c, false, false);
  }

  float* hout = h + n0 + l16;
#pragma unroll
  for (int r = 0; r < 8; ++r)
    hout[(size_t)(m0 + half * 8 + r) * D] = c[r];
}

// ---------------------------------------------------------------------------
// 5) Aggregation: one wave32 per "edge" (E_POS pos edges, then N self loops).
//    Each lane gathers float4 of h[row] (512B coalesced per wave, L2-resident)
//    and atomically adds norm*h into out[col].
// ---------------------------------------------------------------------------
__global__ void aggregate_kernel(const int* __restrict__ pos_ei,
                                 const float* __restrict__ dis,
                                 const float* __restrict__ h,
                                 float* __restrict__ out) {
  const int lane = threadIdx.x & 31;
  const int e    = (blockIdx.x * blockDim.x + threadIdx.x) >> 5;

  int r, c;
  if (e < E_POS) {
    r = pos_ei[e];
    c = pos_ei[E_POS + e];
  } else {
    r = c = e - E_POS;                 // self loop, weight 1
  }
  const float norm = dis[r] * dis[c];

  const v4f hv = *(const v4f*)(h + (size_t)r * D + lane * 4);
  float* op = out + (size_t)c * D + lane * 4;
  atomicAdd(op + 0, norm * hv.x);
  atomicAdd(op + 1, norm * hv.y);
  atomicAdd(op + 2, norm * hv.z);
  atomicAdd(op + 3, norm * hv.w);
}

// ---------------------------------------------------------------------------
// 6) BN statistics: thread t owns channel t; 128-row chunks per block; partial
//    sums merged with float atomics into stats[0..127]=sum, stats[128..255]=sumsq.
// ---------------------------------------------------------------------------
__global__ void stats_kernel(const float* __restrict__ out, float* __restrict__ stats) {
  const int ch = threadIdx.x;          // blockDim.x == 128
  const int r0 = blockIdx.x * 128;
  float s = 0.f, s2 = 0.f;
  for (int i = 0; i < 128; ++i) {
    const int r = r0 + i;
    if (r < N_NODES) {
      const float v = out[(size_t)r * D + ch];
      s  += v;
      s2 += v * v;
    }
  }
  atomicAdd(&stats[ch], s);
  atomicAdd(&stats[D + ch], s2);
}

// ---------------------------------------------------------------------------
// 7) BN finalize + ReLU (in place over out).
// ---------------------------------------------------------------------------
__global__ void bn_relu_kernel(float* __restrict__ out,
                               const float* __restrict__ stats,
                               const float* __restrict__ gamma,
                               const float* __restrict__ beta) {
  const int tid = blockIdx.x * blockDim.x + threadIdx.x;
  if (tid >= N_NODES * D) return;
  const int ch = tid & (D - 1);
  const float invN = 1.0f / (float)N_NODES;
  const float mean = stats[ch] * invN;
  const float var  = stats[D + ch] * invN - mean * mean;
  const float rstd = rsqrtf(var + BN_EPS);
  float v = (out[tid] - mean) * rstd;
  v = gamma[ch] * v + beta[ch];
  out[tid] = v > 0.f ? v : 0.f;
}

// ---------------------------------------------------------------------------
// Launch. Inputs (setup_inputs order): x, pos_edge_index, neg_edge_index,
// W, b, gamma, beta. neg_edge_index is unused (weight 0 => no effect).
// Workspace: h (N*D f32) | deg/dis (N f32) | stats (2*D f32)  ~= 25.8 MB.
// ---------------------------------------------------------------------------
extern "C" void kernel_launch(void* const* d_in, const int* in_sizes, int n_in,
                              void* d_out, int out_size, void* d_ws, size_t ws_size,
                              hipStream_t stream) {
  const float* x     = (const float*)d_in[0];
  const int*   pos   = (const int*)d_in[1];   // int32 per harness convention
  const float* W     = (const float*)d_in[3];
  const float* b     = (const float*)d_in[4];
  const float* gamma = (const float*)d_in[5];
  const float* beta  = (const float*)d_in[6];
  float* out = (float*)d_out;

  char*  ws    = (char*)d_ws;
  float* h     = (float*)ws;                                   // N*D
  float* deg   = (float*)(ws + (size_t)N_NODES * D * sizeof(float)); // N (becomes dis)
  float* stats = deg + N_NODES;                                // 2*D

  init_kernel<<<(N_NODES * D + 255) / 256, 256, 0, stream>>>(out, b, deg, stats);
  deg_kernel<<<(E_POS + 255) / 256, 256, 0, stream>>>(pos, deg);
  rsqrt_kernel<<<(N_NODES + 255) / 256, 256, 0, stream>>>(deg);

  // (N/16)*(D/16) = 25000 tiles, 4 waves per 128-thread block.
  gemm_wmma_kernel<<<(N_NODES / 16) * (D / 16) / 4, 128, 0, stream>>>(x, W, h);

  // (E_POS + N_NODES) edges, 8 waves (edges) per 256-thread block: 68750 blocks.
  aggregate_kernel<<<(E_POS + N_NODES) / 8, 256, 0, stream>>>(pos, deg, h, out);

  stats_kernel<<<(N_NODES + 127) / 128, 128, 0, stream>>>(out, stats);
  bn_relu_kernel<<<(N_NODES * D + 255) / 256, 256, 0, stream>>>(out, stats, gamma, beta);
}